// CostModelV2_34986803593760
// MI455X (gfx1250) — compile-verified
//
#include <hip/hip_runtime.h>

// GINEConv GNN cost model for MI455X (gfx1250, wave32, WMMA).
// All GEMMs use V_WMMA_F32_16X16X4_F32 (full fp32, matches reference precision).
// Working set (~38 MB) is L2-resident (192 MB L2); edge scatter uses f32 atomics.

#define N_NODES   50000
#define N_EDGES   800000
#define NODE_FEAT 176
#define EDGE_FEAT 4
#define HIDDEN    64
#define HID2      128
#define LAYERS    4
#define LN_EPS    1e-5f
#define POOL_BLOCKS 128

typedef __attribute__((ext_vector_type(2))) float v2f;
typedef __attribute__((ext_vector_type(8))) float v8f;

// D = A(16x4) * B(4x16) + C(16x16), all fp32, wave32.
// Lane L (half=L>>4, q=L&15): A frag j = A[q][2*half+j]; B frag j = B[2*half+j][q];
// C/D vgpr j = C[j+8*half][q].
static __device__ __forceinline__ v8f wmma4(v2f a, v2f b, v8f c) {
  return __builtin_amdgcn_wmma_f32_16x16x4_f32(false, a, false, b, (short)0, c,
                                               false, false);
}

// ---------------- input projection: h = x @ in_w + in_b ----------------
__global__ void k_in_proj(const float* __restrict__ x, const float* __restrict__ w,
                          const float* __restrict__ bias, float* __restrict__ h) {
  const int lane = threadIdx.x & 31;
  const int wave = threadIdx.x >> 5;
  const int row0 = (blockIdx.x * 8 + wave) * 16;
  if (row0 >= N_NODES) return;               // no barriers in this kernel
  const int half = lane >> 4;
  const int q    = lane & 15;

  v8f acc[4] = {};
  const float* xrow = x + (size_t)(row0 + q) * NODE_FEAT;

  for (int kk = 0; kk < NODE_FEAT / 4; ++kk) {
    const int k0 = kk * 4 + half * 2;
    v2f a;
    a.x = xrow[k0];
    a.y = xrow[k0 + 1];
#pragma unroll
    for (int t = 0; t < 4; ++t) {
      v2f b;
      b.x = w[(size_t)(k0 + 0) * HIDDEN + t * 16 + q];
      b.y = w[(size_t)(k0 + 1) * HIDDEN + t * 16 + q];
      acc[t] = wmma4(a, b, acc[t]);
    }
  }
#pragma unroll
  for (int t = 0; t < 4; ++t) {
    const float bb = bias[t * 16 + q];
#pragma unroll
    for (int j = 0; j < 8; ++j) {
      const int r = row0 + j + 8 * half;
      h[(size_t)r * HIDDEN + t * 16 + q] = acc[t][j] + bb;
    }
  }
}

// ---------------- zero scratch ----------------
__global__ void k_zero(float4* __restrict__ p, int n4) {
  int i = blockIdx.x * blockDim.x + threadIdx.x;
  const int st = gridDim.x * blockDim.x;
  const float4 z = {0.f, 0.f, 0.f, 0.f};
  for (; i < n4; i += st) p[i] = z;
}

// ------- edge: e = edge_attr @ ew + eb; msg = relu(h[src]+e); aggr[dst] += msg -------
__global__ void k_edge(const float* __restrict__ edge_attr,
                       const long long* __restrict__ ei,  // [2, N_EDGES] int64
                       const float* __restrict__ ew, const float* __restrict__ eb,
                       const float* __restrict__ h, float* __restrict__ aggr) {
  const int lane = threadIdx.x & 31;
  const int wave = threadIdx.x >> 5;
  const int e0   = (blockIdx.x * 8 + wave) * 16;   // grid sized exactly: no tail
  const int half = lane >> 4;
  const int q    = lane & 15;

  // A fragment: 16 edges x 4 edge features (contiguous float2 per lane)
  v2f a;
  a.x = edge_attr[(size_t)(e0 + q) * EDGE_FEAT + 2 * half + 0];
  a.y = edge_attr[(size_t)(e0 + q) * EDGE_FEAT + 2 * half + 1];

  v8f acc[4];
#pragma unroll
  for (int t = 0; t < 4; ++t) {
    v2f b;
    b.x = ew[(2 * half + 0) * HIDDEN + t * 16 + q];
    b.y = ew[(2 * half + 1) * HIDDEN + t * 16 + q];
    v8f z = {};
    acc[t] = wmma4(a, b, z);   // K=4: single WMMA does the whole edge projection
  }

  int s[8], d[8];
#pragma unroll
  for (int j = 0; j < 8; ++j) {
    const int e = e0 + j + 8 * half;
    s[j] = (int)ei[e];
    d[j] = (int)ei[N_EDGES + e];
  }

#pragma unroll
  for (int t = 0; t < 4; ++t) {
    const float bb = eb[t * 16 + q];
#pragma unroll
    for (int j = 0; j < 8; ++j) {
      float msg = acc[t][j] + bb + h[(size_t)s[j] * HIDDEN + t * 16 + q];
      msg = fmaxf(msg, 0.f);
      atomicAdd(&aggr[(size_t)d[j] * HIDDEN + t * 16 + q], msg);   // global_atomic_add_f32
    }
  }
}

// ------- node update: z=h+aggr; t=relu(z@w1+b1); u=t@w2+b2; h=relu(LN(u)) (in place) -------
__global__ void k_node(const float* __restrict__ aggr,
                       const float* __restrict__ w1, const float* __restrict__ b1,
                       const float* __restrict__ w2, const float* __restrict__ b2,
                       const float* __restrict__ lg, const float* __restrict__ lb,
                       float* __restrict__ h) {
  __shared__ __align__(16) float lds[8 * 16 * HID2];   // 64 KB: per-wave 16x128 staging
  const int lane = threadIdx.x & 31;
  const int wave = threadIdx.x >> 5;
  const int row0 = (blockIdx.x * 8 + wave) * 16;
  if (row0 >= N_NODES) return;   // only per-wave LDS + wave_barrier: safe to exit
  const int half = lane >> 4;
  const int q    = lane & 15;
  float* tbuf = lds + wave * 16 * HID2;

  // A fragments of z = h + aggr for all 16 K-steps (K=64)
  v2f afrag[16];
  const float* hrow = h    + (size_t)(row0 + q) * HIDDEN;
  const float* arow = aggr + (size_t)(row0 + q) * HIDDEN;
#pragma unroll
  for (int kk = 0; kk < 16; ++kk) {
    const int k0 = kk * 4 + half * 2;
    afrag[kk].x = hrow[k0]     + arow[k0];
    afrag[kk].y = hrow[k0 + 1] + arow[k0 + 1];
  }

  // GEMM1: t(16x128) = relu(z @ w1 + b1); stream each N-tile through LDS
#pragma unroll
  for (int nt = 0; nt < 8; ++nt) {
    v8f acc = {};
#pragma unroll
    for (int kk = 0; kk < 16; ++kk) {
      const int k0 = kk * 4 + half * 2;
      v2f b;
      b.x = w1[(size_t)(k0 + 0) * HID2 + nt * 16 + q];
      b.y = w1[(size_t)(k0 + 1) * HID2 + nt * 16 + q];
      acc = wmma4(afrag[kk], b, acc);
    }
    const float bb = b1[nt * 16 + q];
#pragma unroll
    for (int j = 0; j < 8; ++j)
      tbuf[(j + 8 * half) * HID2 + nt * 16 + q] = fmaxf(acc[j] + bb, 0.f);
  }

  // same-wave LDS store->load: HW keeps DS ops in order; stop compiler reordering
  __builtin_amdgcn_wave_barrier();

  // GEMM2: u(16x64) = t @ w2  (A fragments re-read from LDS with transposed roles)
  v8f acc2[4] = {};
#pragma unroll
  for (int kk = 0; kk < 32; ++kk) {
    const int k0 = kk * 4 + half * 2;
    v2f a = *(const v2f*)&tbuf[q * HID2 + k0];
#pragma unroll
    for (int nt = 0; nt < 4; ++nt) {
      v2f b;
      b.x = w2[(size_t)(k0 + 0) * HIDDEN + nt * 16 + q];
      b.y = w2[(size_t)(k0 + 1) * HIDDEN + nt * 16 + q];
      acc2[nt] = wmma4(a, b, acc2[nt]);
    }
  }
#pragma unroll
  for (int nt = 0; nt < 4; ++nt) {
    const float bb = b2[nt * 16 + q];
#pragma unroll
    for (int j = 0; j < 8; ++j) acc2[nt][j] += bb;
  }

  // LayerNorm per row: row (j+8*half) lives in the 16 lanes of this half
  float mu[8], rstd[8];
#pragma unroll
  for (int j = 0; j < 8; ++j) {
    float s = 0.f, sq = 0.f;
#pragma unroll
    for (int nt = 0; nt < 4; ++nt) {
      const float v = acc2[nt][j];
      s += v; sq += v * v;
    }
    for (int mk = 1; mk < 16; mk <<= 1) {
      s  += __shfl_xor(s,  mk, 32);
      sq += __shfl_xor(sq, mk, 32);
    }
    const float m = s * (1.f / HIDDEN);
    mu[j]   = m;
    rstd[j] = rsqrtf(sq * (1.f / HIDDEN) - m * m + LN_EPS);
  }
#pragma unroll
  for (int nt = 0; nt < 4; ++nt) {
    const float gg = lg[nt * 16 + q];
    const float bb = lb[nt * 16 + q];
#pragma unroll
    for (int j = 0; j < 8; ++j) {
      const float v = (acc2[nt][j] - mu[j]) * rstd[j] * gg + bb;
      h[(size_t)(row0 + j + 8 * half) * HIDDEN + nt * 16 + q] = fmaxf(v, 0.f);
    }
  }
}

// ---------------- global add pool (partial sums per block) ----------------
__global__ void k_pool(const float* __restrict__ h, float* __restrict__ partial) {
  __shared__ float lds[256];
  const int t = threadIdx.x;
  const int c = t & 63;
  const int rg = t >> 6;                    // 0..3
  float acc = 0.f;
  for (int r = blockIdx.x * 4 + rg; r < N_NODES; r += gridDim.x * 4)
    acc += h[(size_t)r * HIDDEN + c];
  lds[t] = acc;
  __syncthreads();
  if (t < 64)
    partial[blockIdx.x * 64 + t] = lds[t] + lds[t + 64] + lds[t + 128] + lds[t + 192];
}

// ---------------- regressor: out = relu(g@rw1+rb1) @ rw2 + rb2 ----------------
__global__ void k_reg(const float* __restrict__ partial, int nparts,
                      const float* __restrict__ w1, const float* __restrict__ b1,
                      const float* __restrict__ w2, const float* __restrict__ b2,
                      float* __restrict__ out) {
  __shared__ float gs[64];
  const int t = threadIdx.x;               // block of 64 (2 waves)
  float s = 0.f;
  for (int b = 0; b < nparts; ++b) s += partial[b * 64 + t];
  gs[t] = s;
  __syncthreads();
  if (t < 32) {
    float r = b1[t];
    for (int c = 0; c < 64; ++c) r += gs[c] * w1[c * 32 + t];
    r = fmaxf(r, 0.f);
    float v = r * w2[t];
    for (int mk = 16; mk >= 1; mk >>= 1) v += __shfl_xor(v, mk, 32);
    if (t == 0) out[0] = v + b2[0];
  }
}

extern "C" void kernel_launch(void* const* d_in, const int* in_sizes, int n_in,
                              void* d_out, int out_size, void* d_ws, size_t ws_size,
                              hipStream_t stream) {
  (void)in_sizes; (void)n_in; (void)out_size; (void)ws_size;
  const float*     x      = (const float*)d_in[0];
  const long long* ei     = (const long long*)d_in[1];   // int64 edge_index
  const float*     ea     = (const float*)d_in[2];
  const float*     in_w   = (const float*)d_in[3];
  const float*     in_b   = (const float*)d_in[4];
  const float*     edge_w = (const float*)d_in[5];
  const float*     edge_b = (const float*)d_in[6];
  const float*     w1     = (const float*)d_in[7];
  const float*     b1     = (const float*)d_in[8];
  const float*     w2     = (const float*)d_in[9];
  const float*     b2     = (const float*)d_in[10];
  const float*     lg     = (const float*)d_in[11];
  const float*     lb     = (const float*)d_in[12];
  const float*     rw1    = (const float*)d_in[13];
  const float*     rb1    = (const float*)d_in[14];
  const float*     rw2    = (const float*)d_in[15];
  const float*     rb2    = (const float*)d_in[16];
  float* out = (float*)d_out;

  // workspace: h [N,64] (updated in place) | aggr [N,64] | pool partials
  float* h       = (float*)d_ws;
  float* aggr    = h + (size_t)N_NODES * HIDDEN;
  float* partial = aggr + (size_t)N_NODES * HIDDEN;

  const int node_blocks = (N_NODES + 127) / 128;   // 8 waves * 16 rows per block

  k_in_proj<<<node_blocks, 256, 0, stream>>>(x, in_w, in_b, h);
  for (int l = 0; l < LAYERS; ++l) {
    k_zero<<<1024, 256, 0, stream>>>((float4*)aggr, N_NODES * HIDDEN / 4);
    k_edge<<<N_EDGES / 128, 256, 0, stream>>>(ea, ei,
                                              edge_w + (size_t)l * EDGE_FEAT * HIDDEN,
                                              edge_b + (size_t)l * HIDDEN, h, aggr);
    k_node<<<node_blocks, 256, 0, stream>>>(aggr,
                                            w1 + (size_t)l * HIDDEN * HID2,
                                            b1 + (size_t)l * HID2,
                                            w2 + (size_t)l * HID2 * HIDDEN,
                                            b2 + (size_t)l * HIDDEN,
                                            lg + (size_t)l * HIDDEN,
                                            lb + (size_t)l * HIDDEN, h);
  }
  k_pool<<<POOL_BLOCKS, 256, 0, stream>>>(h, partial);
  k_reg<<<1, 64, 0, stream>>>(partial, POOL_BLOCKS, rw1, rb1, rw2, rb2, out);
}